// LocalAttentionBlock_40724879901086
// MI455X (gfx1250) — compile-verified
//
#include <hip/hip_runtime.h>
#include <hip/hip_bf16.h>

#define NSP 64000   // 40*40*40 spatial positions
#define CCH 64      // channels
#define DIMX 40

typedef float v2f __attribute__((ext_vector_type(2)));
typedef float v8f __attribute__((ext_vector_type(8)));

// ---------------------------------------------------------------------------
// Pass 1: q/k/v = x_cl @ W^T + b   (q additionally scaled by hd^-0.5 = 0.25,
// folded into Wq/bq at LDS-fill time).
// x is (C, N) in memory (channels-major), outputs are (N, C) row-major.
// One wave computes a 16-row x 64-col tile for all three matrices using
// V_WMMA_F32_16X16X4_F32, accumulating K=64 in 16 steps of 4.
// ---------------------------------------------------------------------------
__global__ __launch_bounds__(256)
void qkv_gemm_kernel(const float* __restrict__ x,
                     const float* __restrict__ Wq, const float* __restrict__ bq,
                     const float* __restrict__ Wk, const float* __restrict__ bk,
                     const float* __restrict__ Wv, const float* __restrict__ bv,
                     float* __restrict__ qo, float* __restrict__ ko,
                     float* __restrict__ vo)
{
    // B-fragments pre-swizzled so the inner loop ds_load is lane-contiguous:
    // bfrag[((m*4 + t)*16 + kstep)*32 + lane]  -> 3*4*16*32 float2 = 48 KB
    __shared__ v2f bfrag[3 * 4 * 16 * 32];

    const int tid = threadIdx.x;
    for (int e = tid; e < 3 * 4 * 16 * 32; e += 256) {
        const int lane  = e & 31;
        const int kstep = (e >> 5) & 15;
        const int t     = (e >> 9) & 3;
        const int m     = e >> 11;
        const int col   = t * 16 + (lane & 15);            // output channel
        const int krow  = kstep * 4 + ((lane >> 4) << 1);  // input channel
        const float* W  = (m == 0) ? Wq : (m == 1 ? Wk : Wv);
        const float  s  = (m == 0) ? 0.25f : 1.0f;         // fold q scale
        v2f val;
        val.x = W[col * CCH + krow] * s;       // B[k,col] = W[col*64 + k]
        val.y = W[col * CCH + krow + 1] * s;
        bfrag[e] = val;
    }
    __syncthreads();

    const int lane = tid & 31;
    const int wave = tid >> 5;
    const int n0   = blockIdx.x * 128 + wave * 16;  // 500 blocks * 128 rows = 64000
    const int row  = n0 + (lane & 15);
    const int khi  = (lane >> 4) << 1;              // half-wave K offset (0 or 2)

    // A-fragments: 16x4 f32 tile per kstep. Lane m holds row n0+m;
    // VGPR0 = K(kstep*4 + khi), VGPR1 = K(+1).  x is (C,N): coalesced per half-wave.
    v2f a[16];
#pragma unroll
    for (int ks = 0; ks < 16; ++ks) {
        const int ci = ks * 4 + khi;
        a[ks].x = x[(size_t)ci * NSP + row];
        a[ks].y = x[(size_t)(ci + 1) * NSP + row];
    }

#pragma unroll
    for (int m = 0; m < 3; ++m) {
        const float* bias = (m == 0) ? bq : (m == 1 ? bk : bv);
        float*       dst  = (m == 0) ? qo : (m == 1 ? ko : vo);
        const float  bsc  = (m == 0) ? 0.25f : 1.0f;
#pragma unroll
        for (int t = 0; t < 4; ++t) {
            const int   col   = t * 16 + (lane & 15);
            const float binit = bias[col] * bsc;   // same for all 8 acc rows
            v8f acc;
#pragma unroll
            for (int j = 0; j < 8; ++j) acc[j] = binit;
#pragma unroll
            for (int ks = 0; ks < 16; ++ks) {
                const v2f b = bfrag[((m * 4 + t) * 16 + ks) * 32 + lane];
                acc = __builtin_amdgcn_wmma_f32_16x16x4_f32(
                    false, a[ks], false, b, (short)0, acc, false, false);
            }
            // C/D layout: VGPR j -> row M=j (lanes 0-15) / M=j+8 (lanes 16-31)
#pragma unroll
            for (int j = 0; j < 8; ++j) {
                const int r = n0 + j + ((lane >> 4) << 3);
                dst[(size_t)r * CCH + col] = acc[j];
            }
        }
    }
}

// ---------------------------------------------------------------------------
// Pass 2: 3x3x3 local attention + residual, one thread per (position, head).
// Block = 64 consecutive positions x 4 heads -> lane-contiguous C-major I/O.
// q/k/v are (N, 64) from pass 1 (entirely L2-resident: 49 MB < 192 MB).
// ---------------------------------------------------------------------------
__global__ __launch_bounds__(256)
void local_attn_kernel(const float* __restrict__ x,
                       const float* __restrict__ q,
                       const float* __restrict__ k,
                       const float* __restrict__ v,
                       float* __restrict__ out)
{
    const int tid  = threadIdx.x;
    const int nloc = tid & 63;
    const int h    = tid >> 6;
    const int n    = blockIdx.x * 64 + nloc;   // 1000 blocks * 64 = 64000
    const int z    = n / 1600;
    const int rem  = n - z * 1600;
    const int y    = rem / DIMX;
    const int xc   = rem - y * DIMX;
    const int hb   = h * 16;

    float qv[16];
#pragma unroll
    for (int j = 0; j < 4; ++j)
        ((float4*)qv)[j] = ((const float4*)(q + (size_t)n * CCH + hb))[j];

    float wts[27];
    int   nidx[27];
    float wmax = -3.4e38f;
    int c = 0;
#pragma unroll
    for (int dz = -1; dz <= 1; ++dz) {
        const int zz = min(max(z + dz, 0), DIMX - 1);
#pragma unroll
        for (int dy = -1; dy <= 1; ++dy) {
            const int yy = min(max(y + dy, 0), DIMX - 1);
#pragma unroll
            for (int dx = -1; dx <= 1; ++dx) {
                const int xx = min(max(xc + dx, 0), DIMX - 1);
                const int nn = zz * 1600 + yy * DIMX + xx;
                nidx[c] = nn;
                float kbuf[16];
#pragma unroll
                for (int j = 0; j < 4; ++j)
                    ((float4*)kbuf)[j] =
                        ((const float4*)(k + (size_t)nn * CCH + hb))[j];
                float d = 0.0f;
#pragma unroll
                for (int dd = 0; dd < 16; ++dd) d += qv[dd] * kbuf[dd];
                wts[c] = d;
                wmax = fmaxf(wmax, d);
                ++c;
            }
        }
    }

    float sum = 0.0f;
#pragma unroll
    for (int i = 0; i < 27; ++i) {
        const float e = __expf(wts[i] - wmax);
        wts[i] = e;
        sum += e;
    }
    const float inv = 1.0f / sum;

    float acc[16];
#pragma unroll
    for (int dd = 0; dd < 16; ++dd) acc[dd] = 0.0f;
#pragma unroll
    for (int i = 0; i < 27; ++i) {
        const float p = wts[i] * inv;
        float vbuf[16];
#pragma unroll
        for (int j = 0; j < 4; ++j)
            ((float4*)vbuf)[j] =
                ((const float4*)(v + (size_t)nidx[i] * CCH + hb))[j];
#pragma unroll
        for (int dd = 0; dd < 16; ++dd) acc[dd] += p * vbuf[dd];
    }

    // residual + store in (C, N) layout: lanes share h -> consecutive n = coalesced
#pragma unroll
    for (int dd = 0; dd < 16; ++dd) {
        const int cg = hb + dd;
        out[(size_t)cg * NSP + n] = x[(size_t)cg * NSP + n] + acc[dd];
    }
}

extern "C" void kernel_launch(void* const* d_in, const int* in_sizes, int n_in,
                              void* d_out, int out_size, void* d_ws, size_t ws_size,
                              hipStream_t stream) {
    const float* x   = (const float*)d_in[0];
    // d_in[1] = cemb : present in setup_inputs but unused by the reference math
    const float* Wq  = (const float*)d_in[2];
    const float* bq  = (const float*)d_in[3];
    const float* Wk  = (const float*)d_in[4];
    const float* bk  = (const float*)d_in[5];
    const float* Wv  = (const float*)d_in[6];
    const float* bv  = (const float*)d_in[7];
    float* out = (float*)d_out;

    float* qw = (float*)d_ws;                    // 64000*64 f32 = 16.4 MB
    float* kw = qw + (size_t)NSP * CCH;
    float* vw = kw + (size_t)NSP * CCH;          // total 49.2 MB scratch

    qkv_gemm_kernel<<<500, 256, 0, stream>>>(x, Wq, bq, Wk, bk, Wv, bv,
                                             qw, kw, vw);
    local_attn_kernel<<<1000, 256, 0, stream>>>(x, qw, kw, vw, out);
}